// TorchSMoE_SMoE_4355096838868
// MI455X (gfx1250) — compile-verified
//
#include <hip/hip_runtime.h>
#include <cstdint>

// SMoE evaluation: 262144 rows x 28 f32 params -> 262144 x (8x8) f32.
// Bandwidth-bound (~96 MB traffic, ~4.1 us floor @ 23.3 TB/s). No WMMA:
// contractions are K=2/K=4, far below 16x16xK shapes; dominated by exp +
// normalize. CDNA5 path used: async global->LDS staging + s_wait_asynccnt.

#define ROW_F 28            // floats per row
#define ROWS_PER_BLOCK 8    // one row per wave32
#define BLOCK_THREADS 256   // 8 waves

#if defined(__gfx1250__) && __has_builtin(__builtin_amdgcn_global_load_async_to_lds_b128)
#define USE_ASYNC_LDS 1
#else
#define USE_ASYNC_LDS 0
#endif

#if USE_ASYNC_LDS
typedef int v4i __attribute__((vector_size(16)));
typedef __attribute__((address_space(1))) v4i gv4i;   // global (AS1)
typedef __attribute__((address_space(3))) v4i lv4i;   // LDS    (AS3)
#endif

__global__ __launch_bounds__(BLOCK_THREADS)
void smoe_eval_kernel(const float* __restrict__ arr,
                      float* __restrict__ out,
                      int nrows)
{
    __shared__ float sp[ROWS_PER_BLOCK * ROW_F];   // 896 B of staged params

    const int tid  = threadIdx.x;
    const int wave = tid >> 5;
    const int lane = tid & 31;
    const int row0 = blockIdx.x * ROWS_PER_BLOCK;

    // ---- stage 8 rows x 112 B into LDS: 56 lanes x 16 B ----
    const size_t total_f = (size_t)nrows * ROW_F;
    if (tid < (ROWS_PER_BLOCK * ROW_F) / 4) {      // 56 lanes
        const size_t off = (size_t)row0 * ROW_F + (size_t)tid * 4;
        if (off + 4 <= total_f) {
#if USE_ASYNC_LDS
            gv4i* gsrc = (gv4i*)(uintptr_t)(arr + off);
            lv4i* ldst = (lv4i*)(uint32_t)(uintptr_t)(&sp[tid * 4]);
            __builtin_amdgcn_global_load_async_to_lds_b128(gsrc, ldst, 0, 0);
#else
            float4 v = *(const float4*)(arr + off);
            *(float4*)(&sp[tid * 4]) = v;
#endif
        }
    }
#if USE_ASYNC_LDS
#if __has_builtin(__builtin_amdgcn_s_wait_asynccnt)
    __builtin_amdgcn_s_wait_asynccnt(0);
#else
    asm volatile("s_wait_asynccnt 0x0" ::: "memory");
#endif
#endif
    __syncthreads();

    const int row = row0 + wave;
    if (row >= nrows) return;

    // ---- broadcast-read this wave's 28 params from LDS (aligned b128) ----
    const float* p = &sp[wave * ROW_F];
    float4 vcx = *(const float4*)(p + 0);   // center_x[0..3]
    float4 vcy = *(const float4*)(p + 4);   // center_y[0..3]
    float4 vnu = *(const float4*)(p + 8);   // nue_e[0..3]
    float cx[4] = {vcx.x, vcx.y, vcx.z, vcx.w};
    float cy[4] = {vcy.x, vcy.y, vcy.z, vcy.w};
    float nu[4] = {vnu.x, vnu.y, vnu.z, vnu.w};
    float a00[4], a10[4], a11[4];
#pragma unroll
    for (int k = 0; k < 4; ++k) {
        float4 Ak = *(const float4*)(p + 12 + 4 * k);  // [a00, a01(=0 by tril), a10, a11]
        a00[k] = Ak.x;
        a10[k] = Ak.z;
        a11[k] = Ak.w;
    }

    // ---- each lane computes 2 adjacent grid points of the 8x8 domain ----
    // domain index pidx = i*8 + j, x = i/7, y = j/7
    float r2[2];
#pragma unroll
    for (int q = 0; q < 2; ++q) {
        const int pidx = lane * 2 + q;
        const float x = (float)(pidx >> 3) * (1.0f / 7.0f);
        const float y = (float)(pidx & 7)  * (1.0f / 7.0f);
        float num = 0.0f, den = 0.0f;
#pragma unroll
        for (int k = 0; k < 4; ++k) {
            const float d0 = x - cx[k];
            const float d1 = y - cy[k];
            // t = d @ A with A lower-triangular (A01 == 0):
            const float t0 = d0 * a00[k] + d1 * a10[k];
            const float t1 = d1 * a11[k];
            const float e  = __expf(-0.5f * (t0 * t0 + t1 * t1));  // v_exp_f32
            num += e * nu[k];
            den += e;
        }
        den = fmaxf(den, 1e-8f);
        const float r = num * __builtin_amdgcn_rcpf(den);          // v_rcp_f32
        r2[q] = fminf(fmaxf(r, 0.0f), 1.0f);
    }

    // ---- coalesced store: 256 B contiguous per wave (global_store_b64) ----
    float2 v2 = make_float2(r2[0], r2[1]);
    *(float2*)(out + (size_t)row * 64 + (size_t)lane * 2) = v2;
}

extern "C" void kernel_launch(void* const* d_in, const int* in_sizes, int n_in,
                              void* d_out, int out_size, void* d_ws, size_t ws_size,
                              hipStream_t stream)
{
    (void)n_in; (void)out_size; (void)d_ws; (void)ws_size;
    const float* arr = (const float*)d_in[0];
    float* out = (float*)d_out;
    const int nrows = in_sizes[0] / ROW_F;          // 262144
    const int grid  = (nrows + ROWS_PER_BLOCK - 1) / ROWS_PER_BLOCK;
    smoe_eval_kernel<<<grid, BLOCK_THREADS, 0, stream>>>(arr, out, nrows);
}